// Text3DSpaceAll_85306640433866
// MI455X (gfx1250) — compile-verified
//
#include <hip/hip_runtime.h>
#include <hip/hip_bf16.h>
#include <math.h>

// Problem constants (from reference)
#define LW   8
#define GW   17
#define NHW  30               // n_hiper
#define CHW  31               // NH+1 channels
#define DW   768
#define NW   4096
#define NTW  (NW/16)          // 256 scan tiles
#define NCW  (GW*GW*GW)       // 4913 cells

typedef __attribute__((ext_vector_type(2))) float v2f;
typedef __attribute__((ext_vector_type(8))) float v8f;
typedef __attribute__((ext_vector_type(4))) unsigned int v4u;
typedef __attribute__((ext_vector_type(8))) int v8i;
typedef __attribute__((ext_vector_type(4))) int v4i;

__device__ __forceinline__ int clipg(int i) {
    return i < 0 ? 0 : (i > GW - 1 ? GW - 1 : i);
}

// ---------------------------------------------------------------------------
// K1: per-sample tap indices/weights (4 taps per axis, normalized) + dists
// ---------------------------------------------------------------------------
__global__ void k_prep(const float* __restrict__ samples,
                       const float* __restrict__ last_point,
                       float* __restrict__ tapW, int* __restrict__ tapI,
                       float* __restrict__ dist)
{
    int n = blockIdx.x * blockDim.x + threadIdx.x;
    if (n >= NW) return;

    float sx = samples[n*3+0], sy = samples[n*3+1], sz = samples[n*3+2];
    float nx, ny, nz;
    if (n + 1 < NW) { nx = samples[(n+1)*3+0]; ny = samples[(n+1)*3+1]; nz = samples[(n+1)*3+2]; }
    else           { nx = last_point[0];       ny = last_point[1];       nz = last_point[2]; }
    float ex = sx-nx, ey = sy-ny, ez = sz-nz;
    dist[n] = sqrtf(ex*ex + ey*ey + ez*ez);

    float p[3] = { sx + (float)LW, sy + (float)LW, sz + (float)LW };
    #pragma unroll
    for (int ax = 0; ax < 3; ++ax) {
        float pf = floorf(p[ax]);
        float dd = p[ax] - pf;
        int   fi = (int)pf;
        int   ci = (int)ceilf(p[ax]);          // == fi when dd == 0 (y/z use ceil)
        float mlo = (fi > 0)        ? 1.f : 0.f;
        float mhi = (fi + 2 < GW)   ? 1.f : 0.f;
        float inv = 1.f / (2.f + mlo + mhi);
        int i0, i1, i2, i3;
        if (ax == 0) { i0 = clipg(fi); i1 = clipg(fi+1); i2 = clipg(fi-1); i3 = clipg(fi+2); }
        else         { i0 = fi;        i1 = ci;          i2 = clipg(fi-1); i3 = clipg(ci+1); }
        float w0 = (2.f - dd) * inv;
        float w1 = (1.f + dd) * inv;
        float w2 = (1.f - dd) * mlo * inv;
        float w3 = dd * mhi * inv;
        int b = n*12 + ax*4;
        tapW[b+0]=w0; tapW[b+1]=w1; tapW[b+2]=w2; tapW[b+3]=w3;
        tapI[b+0]=i0; tapI[b+1]=i1; tapI[b+2]=i2; tapI[b+3]=i3;
    }
}

// ---------------------------------------------------------------------------
// K2: sd[n,d] = relu(interp of density channel) * dist[n]
//     (density slab = 14.5MB -> L2 resident; 64 taps, coalesced over d)
// ---------------------------------------------------------------------------
__global__ void k_sd(const float* __restrict__ E,
                     const float* __restrict__ tapW, const int* __restrict__ tapI,
                     const float* __restrict__ dist, float* __restrict__ sd)
{
    const int nd = DW / 256;                 // 3
    int n  = blockIdx.x / nd;
    int dc = (blockIdx.x % nd) * 256 + threadIdx.x;
    const float* w = tapW + n*12;
    const int*   I = tapI + n*12;
    float wx[4], wy[4], wz[4]; int ix[4], iy[4], iz[4];
    #pragma unroll
    for (int i = 0; i < 4; ++i) {
        wx[i]=w[i]; wy[i]=w[4+i]; wz[i]=w[8+i];
        ix[i]=I[i]; iy[i]=I[4+i]; iz[i]=I[8+i];
    }
    float acc = 0.f;
    #pragma unroll
    for (int k = 0; k < 4; ++k) {
        if (wz[k] == 0.f) continue;          // uniform per block (same n)
        #pragma unroll
        for (int j = 0; j < 4; ++j) {
            float wzy = wz[k]*wy[j];
            if (wzy == 0.f) continue;
            #pragma unroll
            for (int i = 0; i < 4; ++i) {
                float ww = wzy * wx[i];
                if (ww != 0.f) {
                    long cell = (long)(ix[i]*GW + iy[j])*GW + iz[k];
                    acc += ww * E[cell*(long)(CHW*DW) + (long)NHW*DW + dc];
                }
            }
        }
    }
    float dens = acc > 0.f ? acc : 0.f;
    sd[(long)n*DW + dc] = dens * dist[n];
}

// ---------------------------------------------------------------------------
// K3a: per-16-row tile sums of sd  (fully parallel)
// ---------------------------------------------------------------------------
__global__ void k_tilesum(const float* __restrict__ sd, float* __restrict__ tileS)
{
    const int nd = DW / 256;
    int t  = blockIdx.x / nd;
    int dc = (blockIdx.x % nd) * 256 + threadIdx.x;
    float s = 0.f;
    #pragma unroll
    for (int m = 0; m < 16; ++m) s += sd[(long)(t*16 + m)*DW + dc];
    tileS[(long)t*DW + dc] = s;
}

// ---------------------------------------------------------------------------
// K3b: exclusive scan of the 256 tile sums per d-column (in place).
//      Chunked so 32 loads are in flight before the serial accumulate.
// ---------------------------------------------------------------------------
__global__ void k_tilescan(float* __restrict__ tileS)
{
    int d = blockIdx.x * blockDim.x + threadIdx.x;
    if (d >= DW) return;
    float run = 0.f;
    for (int t0 = 0; t0 < NTW; t0 += 32) {
        float v[32];
        #pragma unroll
        for (int i = 0; i < 32; ++i) v[i] = tileS[(long)(t0+i)*DW + d];
        #pragma unroll
        for (int i = 0; i < 32; ++i) {
            float x = v[i];
            tileS[(long)(t0+i)*DW + d] = run;
            run += x;
        }
    }
}

// ---------------------------------------------------------------------------
// K3c: per-tile inclusive scan on the matrix pipe + weight conversion.
//      D = L(16x16 lower-tri ones) x X via 4 chained V_WMMA_F32_16X16X4_F32.
//      12288 independent waves (one per 16n x 16d tile); sd -> wgt in place.
// ---------------------------------------------------------------------------
__global__ void __launch_bounds__(32) k_wgt(float* __restrict__ sd,
                                            const float* __restrict__ tileS)
{
    const int nd16 = DW / 16;              // 48
    int t    = blockIdx.x / nd16;
    int db   = blockIdx.x % nd16;
    int lane = threadIdx.x;                // 0..31
    int col  = lane & 15;                  // d within tile / M-row for A operand
    int half = lane >> 4;
    int dglob = db*16 + col;
    long rowbase = (long)t * 16;

    float base = tileS[(long)t*DW + dglob];    // exclusive sum of previous tiles

    v8f c = {};
    #pragma unroll
    for (int s = 0; s < 4; ++s) {
        const int k0 = 4*s + 2*half;           // K rows held in this lane's 2 VGPRs
        v2f a, b;
        // A = lower-triangular ones: L[m,k] = (k <= m), m = lane&15
        a.x = (k0     <= col) ? 1.f : 0.f;
        a.y = (k0 + 1 <= col) ? 1.f : 0.f;
        // B rows k0, k0+1 of X[k, d] = sd[rowbase+k, dglob]
        b.x = sd[(rowbase + k0    ) * DW + dglob];
        b.y = sd[(rowbase + k0 + 1) * DW + dglob];
        c = __builtin_amdgcn_wmma_f32_16x16x4_f32(
                false, a, false, b, (short)0, c, false, false);
    }
    #pragma unroll
    for (int r = 0; r < 8; ++r) {
        int  m   = 8*half + r;                 // row of C/D in this lane
        long idx = (rowbase + m) * DW + dglob;
        float x    = sd[idx];
        float incl = c[r];
        float excl = base + incl - x;          // exclusive global cumsum
        float wv   = __expf(-excl) * (1.f - __expf(-x));
        sd[idx] = wv;                          // overwrite sd with wgt
    }
}

// ---------------------------------------------------------------------------
// K4: scatter A[cell,d] += W3[n,t] * wgt[n,d]  (f32 atomics, A = 15MB in L2)
// ---------------------------------------------------------------------------
__global__ void k_scatter(const float* __restrict__ wgt,
                          const float* __restrict__ tapW, const int* __restrict__ tapI,
                          float* __restrict__ A)
{
    const int nd = DW / 256;
    int n  = blockIdx.x / nd;
    int dc = (blockIdx.x % nd) * 256 + threadIdx.x;
    float wv = wgt[(long)n*DW + dc];
    if (wv == 0.f) return;
    const float* w = tapW + n*12;
    const int*   I = tapI + n*12;
    float wx[4], wy[4], wz[4]; int ix[4], iy[4], iz[4];
    #pragma unroll
    for (int i = 0; i < 4; ++i) {
        wx[i]=w[i]; wy[i]=w[4+i]; wz[i]=w[8+i];
        ix[i]=I[i]; iy[i]=I[4+i]; iz[i]=I[8+i];
    }
    #pragma unroll
    for (int k = 0; k < 4; ++k) {
        if (wz[k] == 0.f) continue;
        #pragma unroll
        for (int j = 0; j < 4; ++j) {
            float wzy = wz[k]*wy[j];
            if (wzy == 0.f) continue;
            #pragma unroll
            for (int i = 0; i < 4; ++i) {
                float ww = wzy * wx[i];
                if (ww != 0.f) {
                    long cell = (long)(ix[i]*GW + iy[j])*GW + iz[k];
                    atomicAdd(&A[cell*(long)DW + dc], ww * wv);
                }
            }
        }
    }
}

// ---------------------------------------------------------------------------
// K5: out[h,d] = sum_cell A[cell,d] * E[cell,h,d]
//     Streams the full 468MB embedding table exactly once -> HBM-bound.
//     TDM path: Tensor Data Mover DMA's each 30x256 f32 tile into LDS,
//     double-buffered (s_wait_tensorcnt + barriers) so DMA overlaps FMAs.
// ---------------------------------------------------------------------------
#define CELLS_PER_BLK 64
#define TILE_W 256
#define TILE_H NHW

#if __has_builtin(__builtin_amdgcn_tensor_load_to_lds)
#define USE_TDM 1
#else
#define USE_TDM 0
#endif

#if USE_TDM
// Build a 2D Tensor-DMA descriptor (CDNA5 ISA) and issue the load.
//   group0: [1:0]=count=1, [63:32]=lds byte addr, [120:64]=global addr, [127:126]=type 2
//   group1: data_size=4B; tensor_dim0=768, tensor_dim1=CHW, tile 256x30, dim0 stride=768
// This toolchain exposes the 6-arg builtin:
//   (uint32x4 g0, int32x8 g1, int32x4 g2, int32x4 g3, int32x8 g4, i32 cpol)
__device__ __forceinline__ void tdm_load_tile(unsigned int lds_off, const float* gptr)
{
    unsigned long long ga = (unsigned long long)(const void*)gptr;
    v4u g0;
    g0.x = 0x1u;                                       // count=1, user mode
    g0.y = lds_off;                                    // lds_addr (bytes)
    g0.z = (unsigned int)(ga & 0xffffffffu);           // global_addr[31:0]
    g0.w = (unsigned int)((ga >> 32) & 0x01ffffffu)    // global_addr[56:32]
         | (2u << 30);                                 // type = 2 ("image")
    v8i g1;
    const unsigned int td0 = DW;                       // tensor_dim0
    const unsigned int td1 = CHW;                      // tensor_dim1
    const unsigned int s0  = DW;                       // tensor_dim0_stride
    g1[0] = (int)(2u << 16);                           // data_size=2 (4 bytes), mask=0
    g1[1] = (int)((td0 & 0xffffu) << 16);              // tensor_dim0[15:0]
    g1[2] = (int)(((td0 >> 16) & 0xffffu) | ((td1 & 0xffffu) << 16));
    g1[3] = (int)(((td1 >> 16) & 0xffffu) | ((unsigned)TILE_W << 16)); // tile_dim0
    g1[4] = (int)((unsigned)TILE_H | (0u << 16));      // tile_dim1, tile_dim2=0
    g1[5] = (int)s0;                                   // tensor_dim0_stride[31:0]
    g1[6] = 0;                                         // stride[47:32]=0, dim1_stride lo=0
    g1[7] = 0;                                         // dim1_stride hi=0
    v4i g2 = {0,0,0,0};                                // unused (2D tensor)
    v4i g3 = {0,0,0,0};
    v8i g4 = {0,0,0,0,0,0,0,0};                        // unused extra group
    __builtin_amdgcn_tensor_load_to_lds(g0, g1, g2, g3, g4, 0);
}
#endif

__global__ void k_final(const float* __restrict__ E, const float* __restrict__ A,
                        float* __restrict__ out)
{
    const int nd = DW / TILE_W;            // 3
    int cb = blockIdx.x / nd;
    int dc = (blockIdx.x % nd) * TILE_W + threadIdx.x;
    int c0 = cb * CELLS_PER_BLK;
    int c1 = c0 + CELLS_PER_BLK; if (c1 > NCW) c1 = NCW;
    int num = c1 - c0;

    float acc[NHW];
    #pragma unroll
    for (int h = 0; h < NHW; ++h) acc[h] = 0.f;

#if USE_TDM
    __shared__ float lds_tile[2][TILE_H * TILE_W];     // 2 x 30KB
    const bool w0 = (threadIdx.x < 32);
    const int  d0 = (blockIdx.x % nd) * TILE_W;

    if (w0) {
        tdm_load_tile((unsigned int)(unsigned long long)(void*)&lds_tile[0][0],
                      E + (long)c0*(CHW*DW) + d0);
    }
    for (int i = 0; i < num; ++i) {
        int cur  = i & 1;
        int cell = c0 + i;
        if (w0) __builtin_amdgcn_s_wait_tensorcnt(0);  // tile 'cur' resident
        __syncthreads();
        if (w0 && (i + 1 < num)) {                     // prefetch next while computing
            tdm_load_tile((unsigned int)(unsigned long long)(void*)&lds_tile[cur ^ 1][0],
                          E + (long)(cell + 1)*(CHW*DW) + d0);
        }
        float a = A[(long)cell*DW + dc];
        const float* lt = &lds_tile[cur][0];
        #pragma unroll
        for (int h = 0; h < NHW; ++h)
            acc[h] = fmaf(a, lt[h*TILE_W + threadIdx.x], acc[h]);
        __syncthreads();                               // done reading before overwrite
    }
#else
    for (int cell = c0; cell < c1; ++cell) {
        float a = A[(long)cell*DW + dc];
        const float* eb = E + (long)cell*(CHW*DW) + dc;
        __builtin_prefetch(E + (long)(cell+1)*(CHW*DW) + dc, 0, 0);
        if (a != 0.f) {
            #pragma unroll
            for (int h = 0; h < NHW; ++h) acc[h] = fmaf(a, eb[h*DW], acc[h]);
        }
    }
#endif

    #pragma unroll
    for (int h = 0; h < NHW; ++h) atomicAdd(&out[h*DW + dc], acc[h]);
}

// ---------------------------------------------------------------------------
// K0: zero A (15MB) and out (30x768)
// ---------------------------------------------------------------------------
__global__ void k_zero(float* __restrict__ A, float* __restrict__ out)
{
    long i = (long)blockIdx.x * blockDim.x + threadIdx.x;
    if (i < (long)NCW * DW) A[i] = 0.f;
    if (i < (long)NHW * DW) out[i] = 0.f;
}

// ---------------------------------------------------------------------------
extern "C" void kernel_launch(void* const* d_in, const int* in_sizes, int n_in,
                              void* d_out, int out_size, void* d_ws, size_t ws_size,
                              hipStream_t stream)
{
    const float* samples    = (const float*)d_in[0];   // N x 3
    const float* last_point = (const float*)d_in[1];   // 3
    const float* E          = (const float*)d_in[2];   // G^3 x 31 x 768
    float* out = (float*)d_out;                        // 30 x 768

    // Workspace layout (floats)
    float* ws    = (float*)d_ws;
    float* tapW  = ws;                                 // N*12
    int*   tapI  = (int*)(ws + (long)NW*12);           // N*12
    float* dist  = ws + (long)NW*24;                   // N
    float* sd    = ws + (long)NW*25;                   // N*D  (becomes wgt in-place)
    float* A     = sd + (long)NW*DW;                   // NC*D
    float* tileS = A + (long)NCW*DW;                   // NT*D
    (void)in_sizes; (void)n_in; (void)out_size; (void)ws_size;

    const int nd = DW / 256;                           // 3 d-chunks of 256

    k_zero    <<<(int)(((long)NCW*DW + 255)/256), 256, 0, stream>>>(A, out);
    k_prep    <<<(NW + 255)/256, 256, 0, stream>>>(samples, last_point, tapW, tapI, dist);
    k_sd      <<<NW * nd, 256, 0, stream>>>(E, tapW, tapI, dist, sd);
    k_tilesum <<<NTW * nd, 256, 0, stream>>>(sd, tileS);
    k_tilescan<<<nd, 256, 0, stream>>>(tileS);
    k_wgt     <<<NTW * (DW/16), 32, 0, stream>>>(sd, tileS);
    k_scatter <<<NW * nd, 256, 0, stream>>>(sd, tapW, tapI, A);
    k_final   <<<((NCW + CELLS_PER_BLK - 1)/CELLS_PER_BLK) * nd, 256, 0, stream>>>(E, A, out);
}